// STDDE_45586782879935
// MI455X (gfx1250) — compile-verified
//
#include <hip/hip_runtime.h>
#include <math.h>

typedef __attribute__((ext_vector_type(2))) float v2f;
typedef __attribute__((ext_vector_type(8))) float v8f;

#define NN 1024
#define BB 16
#define HH 64
#define TT 14      // HIST + FUR
#define NK 11
#define CC_ (BB * HH)   // 1024 columns of the state slice

// ------------------------------------------------------------------
// K0: dinv[i] = 1/sqrt(sum_j A[i,j] + 1)
// ------------------------------------------------------------------
__global__ __launch_bounds__(256) void k_dinv(const float* __restrict__ A,
                                              float* __restrict__ dinv) {
  __shared__ float red[256];
  int i = blockIdx.x, t = threadIdx.x;
  const float* row = A + (size_t)i * NN;
  float s = 0.f;
  for (int j = t; j < NN; j += 256) s += row[j];
  red[t] = s; __syncthreads();
  for (int k = 128; k > 0; k >>= 1) { if (t < k) red[t] += red[t + k]; __syncthreads(); }
  if (t == 0) dinv[i] = 1.0f / sqrtf(red[0] + 1.0f);
}

// ------------------------------------------------------------------
// K1: logits = relu(emb @ emb^T)  via V_WMMA_F32_16X16X4_F32
// A frag: lane L -> row tm+L%16, K = k + 2*(L>=16) .. +1   (float2)
// B frag: lane L -> col tn+L%16, K = k + 2*(L>=16) .. +1   (float2, from emb row)
// ------------------------------------------------------------------
__global__ __launch_bounds__(256) void k_semlogits(const float* __restrict__ emb,
                                                   float* __restrict__ out) {
  int wave = threadIdx.x >> 5;
  int lane = threadIdx.x & 31;
  int tile = blockIdx.x * 8 + wave;          // 4096 tiles of 16x16
  int tm = (tile >> 6) << 4;
  int tn = (tile & 63) << 4;
  int half = lane >> 4, l16 = lane & 15;
  v8f acc = {};
  const float* ap = emb + (size_t)(tm + l16) * HH + 2 * half;
  const float* bp = emb + (size_t)(tn + l16) * HH + 2 * half;
#pragma unroll
  for (int k = 0; k < HH; k += 4) {
    v2f a = *(const v2f*)(ap + k);
    v2f b = *(const v2f*)(bp + k);
    acc = __builtin_amdgcn_wmma_f32_16x16x4_f32(false, a, false, b,
                                                (short)0, acc, false, false);
  }
#pragma unroll
  for (int r = 0; r < 8; ++r) {
    float v = acc[r];
    out[(size_t)(tm + r + 8 * half) * NN + tn + l16] = v > 0.f ? v : 0.f;
  }
}

// ------------------------------------------------------------------
// K2: row softmax + spatial + threshold + delay bin (in place)
// ------------------------------------------------------------------
__global__ __launch_bounds__(256) void k_combine(const float* __restrict__ A,
                                                 const float* __restrict__ delay,
                                                 const float* __restrict__ dinv,
                                                 float* __restrict__ wmat,
                                                 float* __restrict__ bmat) {
  __shared__ float red[256];
  int i = blockIdx.x, t = threadIdx.x;
  float lv[4];
#pragma unroll
  for (int q = 0; q < 4; ++q) lv[q] = wmat[(size_t)i * NN + t + 256 * q];
  float m = fmaxf(fmaxf(lv[0], lv[1]), fmaxf(lv[2], lv[3]));
  red[t] = m; __syncthreads();
  for (int k = 128; k > 0; k >>= 1) { if (t < k) red[t] = fmaxf(red[t], red[t + k]); __syncthreads(); }
  m = red[0]; __syncthreads();
  float e[4], s = 0.f;
#pragma unroll
  for (int q = 0; q < 4; ++q) { e[q] = expf(lv[q] - m); s += e[q]; }
  red[t] = s; __syncthreads();
  for (int k = 128; k > 0; k >>= 1) { if (t < k) red[t] += red[t + k]; __syncthreads(); }
  float inv = 1.0f / red[0];
  float di = dinv[i];
#pragma unroll
  for (int q = 0; q < 4; ++q) {
    int j = t + 256 * q;
    float sem = e[q] * inv;
    float ah = A[(size_t)i * NN + j] + (i == j ? 1.0f : 0.0f);
    float ac = sem + di * ah * dinv[j];
    float w = ac > 0.001f ? ac : 0.0f;
    float dl = fminf(fmaxf(delay[(size_t)i * NN + j], 0.0f), 2.0f);
    wmat[(size_t)i * NN + j] = w;
    bmat[(size_t)i * NN + j] = rintf(dl);   // round half-to-even like jnp.round
  }
}

// ------------------------------------------------------------------
// K2T: LDS-tiled transpose + scatter into 3 bin matrices (coalesced writes)
// WT[d][j*N + i] = (bin(i,j)==d) ? W[i,j] : 0
// ------------------------------------------------------------------
__global__ __launch_bounds__(256) void k_binT(const float* __restrict__ wmat,
                                              const float* __restrict__ bmat,
                                              float* __restrict__ WT) {
  __shared__ float lw[32][33];
  __shared__ float lb[32][33];
  int bx = blockIdx.x & 31, by = blockIdx.x >> 5;  // i-tile, j-tile
  int i0 = bx * 32, j0 = by * 32;
  int tx = threadIdx.x & 31, ty = threadIdx.x >> 5;
  for (int r = ty; r < 32; r += 8) {
    lw[r][tx] = wmat[(size_t)(i0 + r) * NN + j0 + tx];
    lb[r][tx] = bmat[(size_t)(i0 + r) * NN + j0 + tx];
  }
  __syncthreads();
  const size_t M1 = (size_t)NN * NN;
  for (int r = ty; r < 32; r += 8) {
    float v = lw[tx][r];
    int bin = (int)lb[tx][r];
    size_t o = (size_t)(j0 + r) * NN + i0 + tx;
    WT[o]          = (bin == 0) ? v : 0.0f;
    WT[o + M1]     = (bin == 1) ? v : 0.0f;
    WT[o + 2 * M1] = (bin == 2) ? v : 0.0f;
  }
}

// ------------------------------------------------------------------
// K3: i2h MLP -> S[:,0..1]; zero S[:,2..13]
// S layout: S[n][t][b][h] at ((n*TT + t)*BB + b)*HH + h
// ------------------------------------------------------------------
__global__ __launch_bounds__(128) void k_initS(const float* __restrict__ ca,
                                               const float* __restrict__ W1,
                                               const float* __restrict__ b1,
                                               const float* __restrict__ W2,
                                               const float* __restrict__ b2,
                                               float* __restrict__ S) {
  __shared__ float mid[64];
  __shared__ float x0[2];
  int n = blockIdx.x >> 4, b = blockIdx.x & 15;
  int t = threadIdx.x;
  if (t < 2) x0[t] = ca[((size_t)(b * NN + n) * NK) * 2 + t];  // knot 0, frac 0
  __syncthreads();
  if (t < 64) {
    float v = x0[0] * W1[t] + x0[1] * W1[64 + t] + b1[t];
    mid[t] = v > 0.f ? v : 0.f;
  }
  __syncthreads();
  float acc = b2[t];
  for (int h = 0; h < 64; ++h) acc += mid[h] * W2[h * 128 + t];
  S[(((size_t)n * TT + (t >> 6)) * BB + b) * HH + (t & 63)] = acc;
  for (int z = t; z < 12 * 64; z += 128) {
    S[(((size_t)n * TT + 2 + (z >> 6)) * BB + b) * HH + (z & 63)] = 0.f;
  }
}

// ------------------------------------------------------------------
// GEMM: agg[i, b*64+h] = sum_d sum_j WT[d][i,j] * S[j, max(idx-d,0), b, h]
// f32 WMMA; each wave computes a 16x32 tile with two accumulators so the
// A-fragment (one b64 load) is reused by two WMMAs -> 2.5 VMEM/WMMA.
// ------------------------------------------------------------------
__global__ __launch_bounds__(256) void k_gemm_agg(const float* __restrict__ WT,
                                                  const float* __restrict__ S,
                                                  float* __restrict__ agg, int idx) {
  int wave = threadIdx.x >> 5;
  int lane = threadIdx.x & 31;
  int tile = blockIdx.x * 8 + wave;        // 2048 tiles of 16x32
  int tm = (tile >> 5) << 4;               // 64 row-tiles
  int tn = (tile & 31) << 5;               // 32 col-groups of width 32
  int half = lane >> 4, l16 = lane & 15;
  v8f acc0 = {}, acc1 = {};
#pragma unroll
  for (int d = 0; d < 3; ++d) {
    int t = idx - d; if (t < 0) t = 0;
    const float* wrow = WT + (size_t)d * NN * NN + (size_t)(tm + l16) * NN + 2 * half;
    // slice row j lives at S + (j*TT + t)*CC_; stepping j adds TT*CC_
    const float* scol = S + ((size_t)(2 * half) * TT + t) * CC_ + tn + l16;
#pragma unroll 4
    for (int k = 0; k < NN; k += 4) {
      v2f a = *(const v2f*)(wrow + k);
      const float* bp = scol + (size_t)k * TT * CC_;
      v2f b0; b0.x = bp[0];  b0.y = bp[TT * CC_];
      v2f b1; b1.x = bp[16]; b1.y = bp[TT * CC_ + 16];
      acc0 = __builtin_amdgcn_wmma_f32_16x16x4_f32(false, a, false, b0,
                                                   (short)0, acc0, false, false);
      acc1 = __builtin_amdgcn_wmma_f32_16x16x4_f32(false, a, false, b1,
                                                   (short)0, acc1, false, false);
    }
  }
#pragma unroll
  for (int r = 0; r < 8; ++r) {
    agg[(size_t)(tm + r + 8 * half) * CC_ + tn + l16]      = acc0[r];
    agg[(size_t)(tm + r + 8 * half) * CC_ + tn + 16 + l16] = acc1[r];
  }
}

// ------------------------------------------------------------------
// Step update: h_new = tanh(agg@Wm + h@Ws + dx@Wx + b)  (STEP==1)
// writes S[:, k+2]; h is S[:, k+1]
// ------------------------------------------------------------------
__global__ __launch_bounds__(64) void k_step(const float* __restrict__ agg,
                                             float* __restrict__ S, int k,
                                             const float* __restrict__ Wm,
                                             const float* __restrict__ Ws,
                                             const float* __restrict__ Wx,
                                             const float* __restrict__ bv,
                                             const float* __restrict__ cb,
                                             const float* __restrict__ cc,
                                             const float* __restrict__ cd) {
  __shared__ float sa[64], sh[64], sdx[2];
  int i = blockIdx.x >> 4, b = blockIdx.x & 15;
  int t = threadIdx.x;
  sa[t] = agg[(size_t)i * CC_ + b * HH + t];
  sh[t] = S[(((size_t)i * TT + (k + 1)) * BB + b) * HH + t];
  if (t < 2) {
    int is = k < 10 ? k : 10;
    float fr = (float)(k - is);
    size_t base = ((size_t)(b * NN + i) * NK + is) * 2 + t;
    sdx[t] = cb[base] + fr * (2.0f * cc[base] + 3.0f * fr * cd[base]);
  }
  __syncthreads();
  float pre = bv[t] + sdx[0] * Wx[t] + sdx[1] * Wx[64 + t];
  for (int h = 0; h < 64; ++h)
    pre += sa[h] * Wm[h * 64 + t] + sh[h] * Ws[h * 64 + t];
  S[(((size_t)i * TT + (k + 2)) * BB + b) * HH + t] = tanhf(pre);
}

// ------------------------------------------------------------------
// Smooth-L1 loss partials (deterministic two-pass reduction)
// ------------------------------------------------------------------
__global__ __launch_bounds__(256) void k_loss(const float* __restrict__ S,
                                              const float* __restrict__ W1,
                                              const float* __restrict__ b1,
                                              const float* __restrict__ W2,
                                              const float* __restrict__ b2,
                                              const float* __restrict__ ca,
                                              const float* __restrict__ cb,
                                              const float* __restrict__ cc,
                                              const float* __restrict__ cd,
                                              float* __restrict__ part) {
  __shared__ float red[256];
  int e = blockIdx.x * 256 + threadIdx.x;   // < 12*1024*16
  int k = e / (NN * BB);
  int r = e % (NN * BB);
  int i = r / BB, b = r % BB;
  const float* yh = S + (((size_t)i * TT + (k + 2)) * BB + b) * HH;
  float y[64];
  for (int h = 0; h < 64; ++h) y[h] = yh[h];
  float reg = b2[0];
  for (int m = 0; m < 32; ++m) {
    float a = b1[m];
    for (int h = 0; h < 64; ++h) a += y[h] * W1[h * 32 + m];
    reg += (a > 0.f ? a : 0.f) * W2[m];
  }
  int is = k < 10 ? k : 10;
  float fr = (float)(k - is);
  size_t base = ((size_t)(b * NN + i) * NK + is) * 2;   // x component 0
  float tgt = ca[base] + fr * (cb[base] + fr * (cc[base] + fr * cd[base]));
  float diff = reg - tgt;
  float ad = fabsf(diff);
  float l = ad < 1.0f ? 0.5f * diff * diff : ad - 0.5f;
  red[threadIdx.x] = l; __syncthreads();
  for (int s = 128; s > 0; s >>= 1) { if (threadIdx.x < s) red[threadIdx.x] += red[threadIdx.x + s]; __syncthreads(); }
  if (threadIdx.x == 0) part[blockIdx.x] = red[0];
}

__global__ __launch_bounds__(256) void k_loss_final(const float* __restrict__ part,
                                                    float* __restrict__ out) {
  __shared__ float red[256];
  int t = threadIdx.x;
  float s = 0.f;
  for (int q = t; q < 768; q += 256) s += part[q];
  red[t] = s; __syncthreads();
  for (int k = 128; k > 0; k >>= 1) { if (t < k) red[t] += red[t + k]; __syncthreads(); }
  if (t == 0) out[196608] = red[0] / 196608.0f;
}

// ------------------------------------------------------------------
// Prediction head: y_pred (B,N,12) from S[:,13]
// ------------------------------------------------------------------
__global__ __launch_bounds__(32) void k_pred(const float* __restrict__ S,
                                             const float* __restrict__ W1,
                                             const float* __restrict__ b1,
                                             const float* __restrict__ W2,
                                             const float* __restrict__ b2,
                                             float* __restrict__ out) {
  __shared__ float mid[32];
  int i = blockIdx.x >> 4, b = blockIdx.x & 15;
  int t = threadIdx.x;
  const float* yl = S + (((size_t)i * TT + 13) * BB + b) * HH;
  float a = b1[t];
  for (int h = 0; h < 64; ++h) a += yl[h] * W1[h * 32 + t];
  mid[t] = a > 0.f ? a : 0.f;
  __syncthreads();
  if (t < 12) {
    float o = b2[t];
    for (int m = 0; m < 32; ++m) o += mid[m] * W2[m * 12 + t];
    out[((size_t)b * NN + i) * 12 + t] = o;
  }
}

// ------------------------------------------------------------------
extern "C" void kernel_launch(void* const* d_in, const int* in_sizes, int n_in,
                              void* d_out, int out_size, void* d_ws, size_t ws_size,
                              hipStream_t stream) {
  (void)in_sizes; (void)n_in; (void)out_size; (void)ws_size;
  const float* A     = (const float*)d_in[0];
  const float* delay = (const float*)d_in[1];
  const float* ca    = (const float*)d_in[2];
  const float* cb    = (const float*)d_in[3];
  const float* cc    = (const float*)d_in[4];
  const float* cd    = (const float*)d_in[5];
  const float* emb   = (const float*)d_in[6];
  const float* i2hW1 = (const float*)d_in[7];
  const float* i2hb1 = (const float*)d_in[8];
  const float* i2hW2 = (const float*)d_in[9];
  const float* i2hb2 = (const float*)d_in[10];
  const float* Wm    = (const float*)d_in[11];
  const float* Wsd   = (const float*)d_in[12];
  const float* Wx    = (const float*)d_in[13];
  const float* db    = (const float*)d_in[14];
  const float* rW1   = (const float*)d_in[15];
  const float* rb1   = (const float*)d_in[16];
  const float* rW2   = (const float*)d_in[17];
  const float* rb2   = (const float*)d_in[18];
  const float* pW1   = (const float*)d_in[19];
  const float* pb1   = (const float*)d_in[20];
  const float* pW2   = (const float*)d_in[21];
  const float* pb2   = (const float*)d_in[22];
  float* out = (float*)d_out;
  float* ws  = (float*)d_ws;

  const size_t M1 = (size_t)NN * NN;        // 1M floats
  float* dinv = ws;                          // 1K
  float* part = ws + 1024;                   // 1K (768 used)
  float* wmat = ws + 8192;                   // 1M: logits -> W -> reused as agg
  float* bmat = wmat + M1;                   // 1M: delay bin ids
  float* WT   = bmat + M1;                   // 3M: transposed bin matrices
  float* S    = WT + 3 * M1;                 // 14M: state (n,t,b,h)
  float* agg  = wmat;                        // reuse after k_binT

  k_dinv<<<NN, 256, 0, stream>>>(A, dinv);
  k_semlogits<<<512, 256, 0, stream>>>(emb, wmat);
  k_combine<<<NN, 256, 0, stream>>>(A, delay, dinv, wmat, bmat);
  k_binT<<<1024, 256, 0, stream>>>(wmat, bmat, WT);
  k_initS<<<NN * BB, 128, 0, stream>>>(ca, i2hW1, i2hb1, i2hW2, i2hb2, S);
  for (int k = 0; k < 12; ++k) {
    k_gemm_agg<<<256, 256, 0, stream>>>(WT, S, agg, 1 + k);
    k_step<<<NN * BB, 64, 0, stream>>>(agg, S, k, Wm, Wsd, Wx, db, cb, cc, cd);
  }
  k_loss<<<768, 256, 0, stream>>>(S, rW1, rb1, rW2, rb2, ca, cb, cc, cd, part);
  k_loss_final<<<1, 256, 0, stream>>>(part, out);
  k_pred<<<NN * BB, 32, 0, stream>>>(S, pW1, pb1, pW2, pb2, out);
}